// TSPGraphEncoder_54357106098197
// MI455X (gfx1250) — compile-verified
//
#include <hip/hip_runtime.h>
#include <hip/hip_bf16.h>

// Problem sizes (fixed by the reference)
#define SS 8
#define BB 16
#define GG 128      // S*B
#define NN 500
#define NP 512      // N padded to multiple of 16 for WMMA tiling
#define DD 128
#define KK 50
#define LL 6
#define MASKW 16    // 512 bits per adjacency row = 16 u32
#define GSTRIDE 516 // gate-strip LDS row stride (516 % 64 == 4 -> bank-conflict free)

typedef __attribute__((ext_vector_type(2))) float    v2f;
typedef __attribute__((ext_vector_type(8))) float    v8f;
typedef __attribute__((ext_vector_type(4))) unsigned v4u;
typedef __attribute__((ext_vector_type(4))) int      v4i;
typedef __attribute__((ext_vector_type(8))) int      v8i;

__device__ __forceinline__ v8f wmma4(v2f a, v2f b, v8f c) {
  // V_WMMA_F32_16X16X4_F32 : D = A(16x4) * B(4x16) + C(16x16), fp32 throughout
  return __builtin_amdgcn_wmma_f32_16x16x4_f32(
      /*neg_a=*/false, a, /*neg_b=*/false, b,
      /*c_mod=*/(short)0, c, /*reuse_a=*/false, /*reuse_b=*/false);
}

// silu via v_exp_f32 + v_rcp_f32 (avoids the correctly-rounded divide chain)
__device__ __forceinline__ float fast_silu(float t) {
  return t * __builtin_amdgcn_rcpf(1.0f + __expf(-t));
}

// ---------------------------------------------------------------- zero mask
__global__ void zero_u32(unsigned* __restrict__ p, int n) {
  int i = blockIdx.x * blockDim.x + threadIdx.x;
  if (i < n) p[i] = 0u;
}

// ---------------------------------------------------------------- kNN graph
// One block per (row i, graph g): build 50-NN edge set, symmetrized via atomicOr.
// Wave-level shuffle argmin; only 2 workgroup barriers per selection round.
__global__ void knn_kernel(const float* __restrict__ x, unsigned* __restrict__ Amask) {
  const int i = blockIdx.x;
  const int g = blockIdx.y;
  const int t = threadIdx.x;              // 128 threads = 4 waves
  const int wave = t >> 5;
  const int lane = t & 31;

  __shared__ float dl[NN];
  __shared__ float wv[4];
  __shared__ int   wi[4];

  const float xi = x[(g * NN + i) * 2 + 0];
  const float yi = x[(g * NN + i) * 2 + 1];

  for (int j = t; j < NN; j += 128) {
    float dx = xi - x[(g * NN + j) * 2 + 0];
    float dy = yi - x[(g * NN + j) * 2 + 1];
    float d = __builtin_amdgcn_sqrtf(dx * dx + dy * dy + 1e-12f);
    dl[j] = (j == i) ? 1e9f : d;          // exclude self (matches +eye*1e9)
  }
  __syncthreads();

  for (int it = 0; it < KK; ++it) {
    float bv = 1e30f; int bi = 0;
    for (int j = t; j < NN; j += 128) {
      float v = dl[j];
      if (v < bv) { bv = v; bi = j; }
    }
    // wave32 shuffle min-reduction (no barriers)
    #pragma unroll
    for (int s = 16; s > 0; s >>= 1) {
      float ov = __shfl_xor(bv, s, 32);
      int   oi = __shfl_xor(bi, s, 32);
      if (ov < bv) { bv = ov; bi = oi; }
    }
    if (lane == 0) { wv[wave] = bv; wi[wave] = bi; }
    __syncthreads();
    if (t == 0) {
      float fv = wv[0]; int fi = wi[0];
      #pragma unroll
      for (int w = 1; w < 4; ++w) if (wv[w] < fv) { fv = wv[w]; fi = wi[w]; }
      atomicOr(&Amask[(g * NP + i ) * MASKW + (fi >> 5)], 1u << (fi & 31));
      atomicOr(&Amask[(g * NP + fi) * MASKW + (i  >> 5)], 1u << (i  & 31));  // A = max(A,A^T)
      dl[fi] = 1e9f;
    }
    __syncthreads();
  }
}

// ---------------------------------------------------------------- h0 encode
__global__ void encode_kernel(const float* __restrict__ x,
                              const float* __restrict__ Wn,
                              const float* __restrict__ bn,
                              float* __restrict__ h) {
  int idx = blockIdx.x * blockDim.x + threadIdx.x;     // G*NP*D threads
  if (idx >= GG * NP * DD) return;
  int g = idx >> 16;           // / (NP*DD)
  int r = (idx >> 7) & (NP - 1);
  int d = idx & (DD - 1);
  float v = 0.0f;
  if (r < NN) {
    float c0 = x[(g * NN + r) * 2 + 0];
    float c1 = x[(g * NN + r) * 2 + 1];
    v = fast_silu(c0 * Wn[d] + c1 * Wn[DD + d] + bn[d]);
  }
  h[idx] = v;
}

// ---------------------------------------------------------------- fused layer
// block = one 16-row tile of one graph; 128 thr = 4 waves, 2 column tiles/wave.
// hT tile is fetched by the Tensor Data Mover (async, overlapped with phases
// 0 and 1); TDM LDS-padding reproduces the 132-float bank-conflict-free stride.
// Phase 0: cooperatively precompute the 16x512 gate strip in LDS.
// Phase 1: m = gate @ h via v_wmma_f32_16x16x4_f32, A-frags from LDS strip.
// Phase 2: h' = silu(h@Ws + m@Wm + bL) via two WMMA chains from LDS tiles.
__global__ void layer_kernel(const float* __restrict__ x,
                             const unsigned* __restrict__ Amask,
                             const float* __restrict__ hin,
                             float* __restrict__ hout,
                             const float* __restrict__ Ws,
                             const float* __restrict__ Wm,
                             const float* __restrict__ bL,
                             const float* __restrict__ we,
                             const float* __restrict__ be,
                             int l) {
  const int i0 = blockIdx.x * 16;
  const int g  = blockIdx.y;
  const int tid   = threadIdx.x;
  const int lane  = tid & 31;
  const int lrow  = lane & 15;
  const int lhalf = lane >> 4;
  const int wave  = tid >> 5;

  __shared__ float cx[NP], cy[NP];
  __shared__ unsigned msk[16 * 17];       // adjacency bits for the 16 tile rows
  __shared__ float hT[16 * 132];          // h tile, filled by TDM with LDS pad
  __shared__ float mT[16 * 132];          // aggregated messages tile
  __shared__ float gT[16 * GSTRIDE];      // gate strip: 16 rows x 512 cols

  // ---- TDM: async DMA of the 16x128 h tile into hT (consumed in phase 2).
  // Descriptor: 2D tile, 4-byte elements, tile 128x16, row stride 128,
  // LDS pad 4 DWORDs every 128 DWORDs -> effective row stride 132 floats.
  if (tid == 0) {
    unsigned lds_off = (unsigned)(unsigned long long)(void*)hT;   // addr[31:0] = LDS offset
    unsigned long long ga =
        (unsigned long long)(const void*)(hin + (size_t)(g * NP + i0) * DD);
    v4u g0;
    g0[0] = 1u;                                     // count=1 valid descriptor
    g0[1] = lds_off;                                // lds_addr
    g0[2] = (unsigned)(ga & 0xFFFFFFFFull);         // global_addr[31:0]
    g0[3] = (unsigned)((ga >> 32) & 0x1FFFFFFull)   // global_addr[56:32]
          | (2u << 30);                             // type = 2 ("image")
    v8i g1;
    g1[0] = (int)((2u << 16)                        // data_size = 4 bytes
          | (1u << 20)                              // pad_enable
          | (6u << 22)                              // pad_interval: 128 DWORDs
          | (3u << 25));                            // pad_amount: 4 DWORDs
    g1[1] = (int)(128u << 16);                      // tensor_dim0[15:0] = 128
    g1[2] = (int)(512u << 16);                      // tensor_dim1[15:0] = 512
    g1[3] = (int)(128u << 16);                      // tile_dim0 = 128
    g1[4] = (int)16;                                // tile_dim1 = 16, tile_dim2 = 0
    g1[5] = (int)128;                               // tensor_dim0_stride = 128
    g1[6] = 0;
    g1[7] = 0;
    v4i gz4 = {0, 0, 0, 0};
    v8i gz8 = {0, 0, 0, 0, 0, 0, 0, 0};
    __builtin_amdgcn_tensor_load_to_lds(g0, g1, gz4, gz4, gz8, 0);
  }

  for (int n = tid; n < NP; n += 128) {
    if (n < NN) {
      cx[n] = x[(g * NN + n) * 2 + 0];
      cy[n] = x[(g * NN + n) * 2 + 1];
    } else { cx[n] = 0.0f; cy[n] = 0.0f; }
  }
  for (int e = tid; e < 16 * MASKW; e += 128) {
    int r = e >> 4, w = e & 15;
    msk[r * 17 + w] = Amask[(g * NP + i0 + r) * MASKW + w];
  }
  __syncthreads();

  const float wel = we[l];
  const float bel = be[l];

  // ---- Phase 0: gate strip (each value computed once per block)
  for (int e = tid; e < 16 * NP; e += 128) {
    const int r = e >> 9;                 // /NP
    const int j = e & (NP - 1);
    const float bit = (float)((msk[r * 17 + (j >> 5)] >> (j & 31)) & 1u);
    const float dx = cx[i0 + r] - cx[j];
    const float dy = cy[i0 + r] - cy[j];
    const float d = __builtin_amdgcn_sqrtf(dx * dx + dy * dy + 1e-12f);
    gT[r * GSTRIDE + j] = bit * fast_silu(wel * d + bel);
  }
  __syncthreads();

  const int c0 = wave * 32;
  const int c1 = c0 + 16;

  // ---- Phase 1: message aggregation m = gate @ h
  v8f m0 = {0,0,0,0,0,0,0,0};
  v8f m1 = {0,0,0,0,0,0,0,0};
  for (int j = 0; j < NP; j += 4) {
    const int k0 = j + (lhalf << 1);
    v2f a = {gT[lrow * GSTRIDE + k0], gT[lrow * GSTRIDE + k0 + 1]};
    const float* hr0 = hin + (size_t)(g * NP + k0) * DD;
    v2f b0 = {hr0[c0 + lrow], hr0[DD + c0 + lrow]};
    v2f b1 = {hr0[c1 + lrow], hr0[DD + c1 + lrow]};
    m0 = wmma4(a, b0, m0);
    m1 = wmma4(a, b1, m1);
  }

  // stage m tile to LDS (C/D layout: M = r + 8*lhalf, N = lane%16)
  #pragma unroll
  for (int r = 0; r < 8; ++r) {
    mT[(r + 8 * lhalf) * 132 + c0 + lrow] = m0[r];
    mT[(r + 8 * lhalf) * 132 + c1 + lrow] = m1[r];
  }
  // TDM tile must be complete before any wave reads hT after this barrier.
  if (tid == 0) __builtin_amdgcn_s_wait_tensorcnt(0);
  __syncthreads();

  // ---- Phase 2: h' = silu(h@Ws + m@Wm + bL)
  const float* WsL = Ws + (size_t)l * DD * DD;
  const float* WmL = Wm + (size_t)l * DD * DD;
  const float bias0 = bL[l * DD + c0 + lrow];
  const float bias1 = bL[l * DD + c1 + lrow];
  v8f d0 = {bias0, bias0, bias0, bias0, bias0, bias0, bias0, bias0};
  v8f d1 = {bias1, bias1, bias1, bias1, bias1, bias1, bias1, bias1};

  for (int k = 0; k < DD; k += 4) {
    const int kk = k + (lhalf << 1);
    v2f ah = {hT[lrow * 132 + kk], hT[lrow * 132 + kk + 1]};
    v2f am = {mT[lrow * 132 + kk], mT[lrow * 132 + kk + 1]};
    v2f bs0 = {WsL[kk * DD + c0 + lrow], WsL[(kk + 1) * DD + c0 + lrow]};
    v2f bm0 = {WmL[kk * DD + c0 + lrow], WmL[(kk + 1) * DD + c0 + lrow]};
    v2f bs1 = {WsL[kk * DD + c1 + lrow], WsL[(kk + 1) * DD + c1 + lrow]};
    v2f bm1 = {WmL[kk * DD + c1 + lrow], WmL[(kk + 1) * DD + c1 + lrow]};
    d0 = wmma4(ah, bs0, d0);
    d0 = wmma4(am, bm0, d0);
    d1 = wmma4(ah, bs1, d1);
    d1 = wmma4(am, bm1, d1);
  }

  #pragma unroll
  for (int r = 0; r < 8; ++r) {
    const int row = i0 + r + 8 * lhalf;
    hout[(size_t)(g * NP + row) * DD + c0 + lrow] = fast_silu(d0[r]);
    hout[(size_t)(g * NP + row) * DD + c1 + lrow] = fast_silu(d1[r]);
  }
}

// ---------------------------------------------------------------- pool + Wo
__global__ void pool_kernel(const float* __restrict__ h,
                            const float* __restrict__ Wo,
                            float* __restrict__ out) {
  const int g = blockIdx.x;
  const int t = threadIdx.x;              // 128 threads = D
  float s = 0.0f;
  for (int n = 0; n < NN; ++n) s += h[(size_t)(g * NP + n) * DD + t];
  __shared__ float mean[DD];
  mean[t] = s * (1.0f / (float)NN);
  __syncthreads();
  float o = 0.0f;
  for (int d = 0; d < DD; ++d) o += mean[d] * Wo[d * DD + t];
  out[g * DD + t] = o;
}

// ---------------------------------------------------------------- launcher
extern "C" void kernel_launch(void* const* d_in, const int* in_sizes, int n_in,
                              void* d_out, int out_size, void* d_ws, size_t ws_size,
                              hipStream_t stream) {
  const float* x  = (const float*)d_in[0];   // (S,B,N,2)
  const float* Wn = (const float*)d_in[1];   // (2,D)
  const float* bn = (const float*)d_in[2];   // (D)
  const float* Ws = (const float*)d_in[3];   // (L,D,D)
  const float* Wm = (const float*)d_in[4];   // (L,D,D)
  const float* bL = (const float*)d_in[5];   // (L,D)
  const float* we = (const float*)d_in[6];   // (L)
  const float* be = (const float*)d_in[7];   // (L)
  const float* Wo = (const float*)d_in[8];   // (D,D)
  float* out = (float*)d_out;

  char* ws = (char*)d_ws;
  unsigned* Amask = (unsigned*)ws;                                  // G*NP*16 u32 = 4 MB
  float* hA = (float*)(ws + (size_t)GG * NP * MASKW * 4);           // G*NP*D f32 = 32 MB
  float* hB = hA + (size_t)GG * NP * DD;

  // 1) clear adjacency bitmask (atomicOr accumulates; must start from zero)
  {
    int n = GG * NP * MASKW;
    zero_u32<<<(n + 255) / 256, 256, 0, stream>>>(Amask, n);
  }
  // 2) kNN edge set
  knn_kernel<<<dim3(NN, GG), 128, 0, stream>>>(x, Amask);
  // 3) node encoder
  {
    int n = GG * NP * DD;
    encode_kernel<<<(n + 255) / 256, 256, 0, stream>>>(x, Wn, bn, hA);
  }
  // 4) six fused message-passing layers (ping-pong)
  float* cur = hA;
  float* nxt = hB;
  for (int l = 0; l < LL; ++l) {
    layer_kernel<<<dim3(NP / 16, GG), 128, 0, stream>>>(
        x, Amask, cur, nxt, Ws, Wm, bL, we, be, l);
    float* tmp = cur; cur = nxt; nxt = tmp;
  }
  // 5) global mean pool + output projection (cur == hA after 6 swaps)
  pool_kernel<<<GG, 128, 0, stream>>>(cur, Wo, out);
}